// SpatialPoolingLayer_60799557042177
// MI455X (gfx1250) — compile-verified
//
#include <hip/hip_runtime.h>

// ---------------------------------------------------------------------------
// HTM Spatial Pooler step, MI455X (gfx1250, wave32)
//   x : [1, 8192]  f32
//   p : [8192, 8192] f32   (row-major, OUT fastest)
//   out: y [8192] f32  ++  p_new [8192*8192] f32
// ---------------------------------------------------------------------------

typedef __attribute__((ext_vector_type(2))) float v2f;
typedef __attribute__((ext_vector_type(8))) float v8f;
typedef __attribute__((ext_vector_type(4))) float f4;

static constexpr int IN   = 8192;
static constexpr int OUT  = 8192;
static constexpr int KWIN = 164;          // round(0.02 * 8192)
static constexpr float LR = 0.1f;

static constexpr int ROWSEG       = 16;          // row segments for overlap
static constexpr int ROWS_PER_SEG = IN / ROWSEG; // 512
static constexpr int WAVES_PER_BLK = 8;          // 256 threads / 32

// ---------------------------------------------------------------------------
// Kernel 1: partial overlap via V_WMMA_F32_16X16X4_F32 (exact f32 GEMV).
// One wave handles one 16-column group over one 512-row segment.
//   A (16x4 f32, 2 VGPR):  lane0  -> {x[k],   x[k+1]}  (M=0, K0/K1)
//                          lane16 -> {x[k+2], x[k+3]}  (M=0, K2/K3)
//                          all other lanes 0 (rows M=1..15 are zero)
//   B (4x16 f32, 2 VGPR):  lane n   (0..15) -> {c[k  ][col+n], c[k+1][col+n]}
//                          lane n+16        -> {c[k+2][col+n], c[k+3][col+n]}
//   D row 0 (VGPR0, lanes 0..15) accumulates overlap for the 16 columns.
// Branch-free A construction: A = msel * {x[k+2*sub], x[k+1+2*sub]}, where
// msel is 1.0 only on lanes 0 and 16 (the M=0 slots of each half-wave).
// ---------------------------------------------------------------------------
__global__ void overlap_wmma_kernel(const float* __restrict__ x,
                                    const float* __restrict__ p,
                                    float* __restrict__ partial) {
  const int lane   = threadIdx.x & 31;
  const int wave   = threadIdx.x >> 5;
  const int colgrp = blockIdx.x * WAVES_PER_BLK + wave;   // 0..511
  const int seg    = blockIdx.y;                          // 0..15
  const int col0   = colgrp * 16;
  const int row0   = seg * ROWS_PER_SEG;

  const int sub    = lane >> 4;      // 0: rows k,k+1   1: rows k+2,k+3
  const int nlane  = lane & 15;      // column within group

  const float msel = (nlane == 0) ? 1.0f : 0.0f;  // lanes 0 and 16 carry M=0

  const float* pb = p + (size_t)(row0 + sub * 2) * OUT + col0 + nlane;
  const float* xb = x + row0 + sub * 2;           // 8-byte aligned for k%4==0

  v8f acc = {};
  for (int k = 0; k < ROWS_PER_SEG; k += 4) {
    // connected-synapse bits (exact {0,1} in f32)
    const float b0 = (pb[(size_t)(k    ) * OUT] > 0.5f) ? 1.0f : 0.0f;
    const float b1 = (pb[(size_t)(k + 1) * OUT] > 0.5f) ? 1.0f : 0.0f;

    const v2f xv = *(const v2f*)(xb + k);         // {x[k+2s], x[k+1+2s]}

    v2f A, B;
    A[0] = msel * xv[0];
    A[1] = msel * xv[1];
    B[0] = b0;
    B[1] = b1;

    acc = __builtin_amdgcn_wmma_f32_16x16x4_f32(
        /*neg_a=*/false, A, /*neg_b=*/false, B,
        /*c_mod=*/(short)0, acc, /*reuse_a=*/false, /*reuse_b=*/false);
  }

  // D layout: VGPR0, lanes 0..15 = row M=0, N=lane
  if (lane < 16) {
    partial[(size_t)seg * OUT + col0 + lane] = acc[0];
  }
}

// ---------------------------------------------------------------------------
// Kernel 2: deterministic fixed-order reduction of row-segment partials.
// ---------------------------------------------------------------------------
__global__ void reduce_partials_kernel(const float* __restrict__ partial,
                                       float* __restrict__ overlap) {
  const int j = blockIdx.x * blockDim.x + threadIdx.x;
  float s = 0.0f;
  #pragma unroll
  for (int r = 0; r < ROWSEG; ++r) s += partial[(size_t)r * OUT + j];
  overlap[j] = s;
}

// ---------------------------------------------------------------------------
// Kernel 3: k-winner-take-all by rank counting (tie-exact, deterministic).
//   y[j] = ( #{ i : overlap[i] > overlap[j] } < K )  <=>  overlap[j] >= kth
// ---------------------------------------------------------------------------
__global__ void kwta_kernel(const float* __restrict__ overlap,
                            float* __restrict__ y_out) {
  __shared__ float ov[OUT];                       // 32 KB of 320 KB
  for (int i = threadIdx.x; i < OUT; i += blockDim.x) ov[i] = overlap[i];
  __syncthreads();

  const int j = blockIdx.x * blockDim.x + threadIdx.x;
  const float v = ov[j];
  int cnt = 0;
  for (int i = 0; i < OUT; ++i) cnt += (ov[i] > v) ? 1 : 0;
  y_out[j] = (cnt < KWIN) ? 1.0f : 0.0f;
}

// ---------------------------------------------------------------------------
// Kernel 4: streaming Hebbian update, float4 + nontemporal (256MB > 192MB L2).
//   c     = p > 0.5
//   delta = LR * ((c*x_i - 0.5) * 2) * y_j
//   p_new = max(p + delta, -1)
// ---------------------------------------------------------------------------
__global__ void update_kernel(const float* __restrict__ x,
                              const float* __restrict__ p,
                              const float* __restrict__ y,
                              float* __restrict__ p_out) {
  const size_t idx = ((size_t)blockIdx.x * blockDim.x + threadIdx.x) * 4;
  const int i = (int)(idx / OUT);     // 4 elems never straddle a row (OUT%4==0)
  const int j = (int)(idx % OUT);

  const float xi = x[i];
  const f4 pv = __builtin_nontemporal_load((const f4*)(p + idx));
  const f4 yv = *(const f4*)(y + j);  // 32 KB, stays cached

  f4 r;
  #pragma unroll
  for (int e = 0; e < 4; ++e) {
    const float pe = pv[e];
    const float c  = (pe > 0.5f) ? 1.0f : 0.0f;
    const float al = (c * xi - 0.5f) * 2.0f;
    const float d  = LR * al * yv[e];
    r[e] = fmaxf(pe + d, -1.0f);
  }
  __builtin_nontemporal_store(r, (f4*)(p_out + idx));
}

// ---------------------------------------------------------------------------
extern "C" void kernel_launch(void* const* d_in, const int* in_sizes, int n_in,
                              void* d_out, int out_size, void* d_ws, size_t ws_size,
                              hipStream_t stream) {
  const float* x = (const float*)d_in[0];   // [1, 8192]
  const float* p = (const float*)d_in[1];   // [8192, 8192]

  float* y_out = (float*)d_out;             // first 8192 floats
  float* p_out = (float*)d_out + OUT;       // next 8192*8192 floats

  float* partial = (float*)d_ws;                    // ROWSEG*OUT f32 (512 KB)
  float* overlap = partial + (size_t)ROWSEG * OUT;  // OUT f32     (32 KB)

  // 1) overlap partials: (512 col-groups / 8 waves) x 16 row segments
  dim3 g1(OUT / 16 / WAVES_PER_BLK, ROWSEG);        // (64, 16)
  overlap_wmma_kernel<<<g1, WAVES_PER_BLK * 32, 0, stream>>>(x, p, partial);

  // 2) reduce partials -> overlap
  reduce_partials_kernel<<<OUT / 256, 256, 0, stream>>>(partial, overlap);

  // 3) k-WTA -> y
  kwta_kernel<<<OUT / 256, 256, 0, stream>>>(overlap, y_out);

  // 4) permanence update -> p_new
  const size_t total = (size_t)IN * OUT;
  update_kernel<<<(unsigned)(total / 4 / 256), 256, 0, stream>>>(x, p, y_out, p_out);
}